// DeepGraphConvolution_90031104459405
// MI455X (gfx1250) — compile-verified
//
#include <hip/hip_runtime.h>

#define NNODES 100000
#define NEDGES 1600000
#define FDIM 128

typedef float v2f __attribute__((ext_vector_type(2)));
typedef float v8f __attribute__((ext_vector_type(8)));

// Non-returning hardware f32 atomic add (CDNA5 GLOBAL_ATOMIC_ADD_F32).
// Bypasses compiler fp-atomic expansion policy; tracked by STOREcnt and
// drained by the implicit wait at S_ENDPGM.
__device__ __forceinline__ void global_fadd(float* p, float v) {
    asm volatile("global_atomic_add_f32 %0, %1, off" : : "v"(p), "v"(v) : "memory");
}

// ---------------------------------------------------------------------------
// GEMM: Y[N,128] = (RELU ? relu(X) : X)[N,128] @ W[128,128]
// One wave32 computes a 16-row strip x all 128 columns using
// V_WMMA_F32_16X16X4_F32 (exact fp32).
// W is staged in LDS pair-interleaved: lw[(k>>1)*256 + col*2 + (k&1)],
// so each B fragment {W[k][col], W[k+1][col]} is one aligned ds_load_b64.
// ---------------------------------------------------------------------------
template <bool RELU>
__global__ __launch_bounds__(256) void gemm128_wmma_f32(
    const float* __restrict__ X, const float* __restrict__ W,
    float* __restrict__ Y, int n_strips)
{
    __shared__ float lw[FDIM * FDIM];  // 64 KB
    const int tid = threadIdx.x;
    {
        const float4* W4 = (const float4*)W;
        #pragma unroll
        for (int i = 0; i < 16; ++i) {             // 256 thr * 16 * 16B = 64 KB
            float4 v = W4[tid + i * 256];
            const int linear = (tid + i * 256) * 4; // element index of v.x
            const int k = linear >> 7;              // row of W
            const int n = linear & 127;             // col of W
            float* d = &lw[((k >> 1) << 8) + (n << 1) + (k & 1)];
            d[0] = v.x; d[2] = v.y; d[4] = v.z; d[6] = v.w;
        }
    }
    __syncthreads();

    const int wave = (blockIdx.x * 256 + tid) >> 5;  // uniform per wave
    if (wave >= n_strips) return;
    const int lane = tid & 31;
    const int half = lane >> 4;   // 0: K={k,k+1}, 1: K={k+2,k+3}
    const int lm   = lane & 15;   // row within strip (A) / col within tile (B,C)

    // A fragment source: row (strip*16 + lm), starting col 2*half
    const float* xrow = X + (size_t)(wave * 16 + lm) * FDIM + 2 * half;

    v8f acc[8] = {};  // 8 N-tiles of 16x16 f32 accumulators

    for (int k = 0; k < FDIM; k += 4) {
        float a0 = xrow[k];
        float a1 = xrow[k + 1];
        if (RELU) { a0 = fmaxf(a0, 0.f); a1 = fmaxf(a1, 0.f); }
        v2f a; a.x = a0; a.y = a1;
        // Pair row (k + 2*half)>>1 holds {W[k_eff][*], W[k_eff+1][*]} interleaved
        const float* bp = &lw[(((k + 2 * half) >> 1) << 8) + (lm << 1)];
        #pragma unroll
        for (int n = 0; n < 8; ++n) {
            v2f b = *(const v2f*)(bp + n * 32);     // one ds_load_b64
            acc[n] = __builtin_amdgcn_wmma_f32_16x16x4_f32(
                false, a, false, b, (short)0, acc[n], false, false);
        }
    }

    // C/D layout: VGPR v, lanes 0-15 -> M=v, lanes 16-31 -> M=v+8; N = lm
    float* yb = Y + (size_t)wave * 16 * FDIM;
    #pragma unroll
    for (int n = 0; n < 8; ++n) {
        #pragma unroll
        for (int v = 0; v < 8; ++v) {
            yb[(size_t)(v + 8 * half) * FDIM + n * 16 + lm] = acc[n][v];
        }
    }
}

// ---------------------------------------------------------------------------
// SpMM scatter: O[row] += w * X[col] per edge. One wave32 per edge; lane l
// owns features [4l, 4l+3] (512 B coalesced gather). Atomics land in L2.
// ---------------------------------------------------------------------------
__global__ __launch_bounds__(256) void spmm_scatter_f32(
    const int* __restrict__ erow, const int* __restrict__ ecol,
    const float* __restrict__ ew, const float* __restrict__ X,
    float* __restrict__ O, int n_edges)
{
    const int t = blockIdx.x * 256 + threadIdx.x;
    const int lane = threadIdx.x & 31;
    const int wave = t >> 5;
    const int n_waves = (gridDim.x * 256) >> 5;
    for (int e = wave; e < n_edges; e += n_waves) {
        const int r = erow[e];
        const int c = ecol[e];
        const float wt = ew[e];
        float4 v = ((const float4*)(X + (size_t)c * FDIM))[lane];
        float* ob = O + (size_t)r * FDIM + lane * 4;
        global_fadd(ob + 0, wt * v.x);
        global_fadd(ob + 1, wt * v.y);
        global_fadd(ob + 2, wt * v.z);
        global_fadd(ob + 3, wt * v.w);
    }
}

// ---------------------------------------------------------------------------
// Utility kernels
// ---------------------------------------------------------------------------
__global__ __launch_bounds__(256) void zero_f32(float4* __restrict__ p, int n4)
{
    int i = blockIdx.x * 256 + threadIdx.x;
    const int stride = gridDim.x * 256;
    const float4 z = {0.f, 0.f, 0.f, 0.f};
    for (; i < n4; i += stride) p[i] = z;
}

__global__ __launch_bounds__(256) void relu_f32(float4* __restrict__ p, int n4)
{
    int i = blockIdx.x * 256 + threadIdx.x;
    const int stride = gridDim.x * 256;
    for (; i < n4; i += stride) {
        float4 v = p[i];
        v.x = fmaxf(v.x, 0.f); v.y = fmaxf(v.y, 0.f);
        v.z = fmaxf(v.z, 0.f); v.w = fmaxf(v.w, 0.f);
        p[i] = v;
    }
}

// ---------------------------------------------------------------------------
// Orchestration:
//   L1: ws0 = x  @ w1 ; ws1 = 0 ; ws1 += spmm(ws0)
//   L2: ws0 = relu(ws1) @ w2 ; ws1 = 0 ; ws1 += spmm(ws0)
//   L3: ws0 = relu(ws1) @ w3 ; out = 0 ; out += spmm(ws0) ; out = relu(out)
// ---------------------------------------------------------------------------
extern "C" void kernel_launch(void* const* d_in, const int* in_sizes, int n_in,
                              void* d_out, int out_size, void* d_ws, size_t ws_size,
                              hipStream_t stream)
{
    const float* x    = (const float*)d_in[0];
    const int*   erow = (const int*)  d_in[1];
    const int*   ecol = (const int*)  d_in[2];
    const float* ew   = (const float*)d_in[3];
    const float* w1   = (const float*)d_in[4];
    const float* w2   = (const float*)d_in[5];
    const float* w3   = (const float*)d_in[6];
    float* out = (float*)d_out;

    float* ws0 = (float*)d_ws;                     // GEMM output
    float* ws1 = ws0 + (size_t)NNODES * FDIM;      // SpMM accumulator

    const int n_strips   = NNODES / 16;            // 6250 (exact)
    const int gemm_grid  = (n_strips + 7) / 8;     // 8 waves per 256-thr block
    const int spmm_grid  = 4096;                   // 32768 waves, ~49 edges each
    const int n4         = (NNODES * FDIM) / 4;    // 3.2M float4
    const int util_grid  = 2048;

    // Layer 1
    gemm128_wmma_f32<false><<<gemm_grid, 256, 0, stream>>>(x,   w1, ws0, n_strips);
    zero_f32              <<<util_grid, 256, 0, stream>>>((float4*)ws1, n4);
    spmm_scatter_f32      <<<spmm_grid, 256, 0, stream>>>(erow, ecol, ew, ws0, ws1, NEDGES);

    // Layer 2 (relu of ws1 fused into GEMM A-fragment load)
    gemm128_wmma_f32<true><<<gemm_grid, 256, 0, stream>>>(ws1, w2, ws0, n_strips);
    zero_f32              <<<util_grid, 256, 0, stream>>>((float4*)ws1, n4);
    spmm_scatter_f32      <<<spmm_grid, 256, 0, stream>>>(erow, ecol, ew, ws0, ws1, NEDGES);

    // Layer 3
    gemm128_wmma_f32<true><<<gemm_grid, 256, 0, stream>>>(ws1, w3, ws0, n_strips);
    zero_f32              <<<util_grid, 256, 0, stream>>>((float4*)out, n4);
    spmm_scatter_f32      <<<spmm_grid, 256, 0, stream>>>(erow, ecol, ew, ws0, out, NEDGES);
    relu_f32              <<<util_grid, 256, 0, stream>>>((float4*)out, n4);
}